// SimpleVQVAE_27212912787878
// MI455X (gfx1250) — compile-verified
//
#include <hip/hip_runtime.h>

// ---------------------------------------------------------------------------
// SimpleVQVAE forward on gfx1250 (MI455X), fp32 WMMA (v_wmma_f32_16x16x4_f32)
//
// B=16 T=2048 IN=80 H=512 D=512 K=1024, M = B*T = 32768
// conv1d(K=3, SAME) lowered to 3 shifted GEMM accumulation passes.
// VQ distance = GEMM vs codebook^T + lane-shuffle argmin reduction.
//
// - LDS tiles and the pre-transformed B operands (weights / codebook^T) are
//   stored as packed k-pairs (v2f): every WMMA fragment is one b64 load.
// - Double-buffered staging: one barrier per K-step; next tile's global
//   loads overlap the current tile's 32 WMMAs.
// - All 8 B fragments of a k-step are loaded into distinct registers before
//   the WMMA burst so DS latency overlaps the matrix pipe.
// - KC / NN are template constants: no runtime division / 64-bit muls in the
//   hot loop.
// ---------------------------------------------------------------------------

typedef __attribute__((ext_vector_type(2))) float v2f;
typedef __attribute__((ext_vector_type(8))) float v8f;

#define TSEQ  2048
#define TMASK 2047
#define BM 128
#define BN 128
#define BK 16

__device__ __forceinline__ v8f wmma_f32x4(v2f a, v2f b, v8f c) {
  // 8 args: (neg_a, A, neg_b, B, c_mod, C, reuse_a, reuse_b)
  return __builtin_amdgcn_wmma_f32_16x16x4_f32(false, a, false, b, (short)0, c,
                                               false, false);
}

// ---------------------------------------------------------------------------
// Weight re-layout: w[Cout][Cin][3] (torch OIH) -> pair-packed
//   logical W[kdim][o], kdim = kk*Cin + i
//   packed  Wp[(kdim>>1)*(2*Cout) + 2*o + (kdim&1)]
// ---------------------------------------------------------------------------
__global__ void vqvae_wtrans(const float* __restrict__ w, float* __restrict__ wp,
                             int Cout, int Cin) {
  int t = blockIdx.x * blockDim.x + threadIdx.x;
  int total = Cout * Cin * 3;
  if (t < total) {
    int o    = t / (Cin * 3);
    int rem  = t - o * (Cin * 3);
    int i    = rem / 3;
    int kk   = rem - i * 3;
    int kdim = kk * Cin + i;
    wp[(long)(kdim >> 1) * (2 * Cout) + 2 * o + (kdim & 1)] = w[t];
  }
}

// codebook [K][D] -> pair-packed transpose:
//   logical CBT[d][k] = cb[k][d];  packed CBTp[(d>>1)*(2*K) + 2*k + (d&1)]
__global__ void vqvae_cbT(const float* __restrict__ cb, float* __restrict__ cbp,
                          int K, int D) {
  int t = blockIdx.x * blockDim.x + threadIdx.x;
  if (t < K * D) {
    int k = t / D;
    int d = t - k * D;
    cbp[(long)(d >> 1) * (2 * K) + 2 * k + (d & 1)] = cb[t];
  }
}

// cnorm[k] = sum_d codebook[k][d]^2
__global__ void vqvae_cnorm(const float* __restrict__ cb, float* __restrict__ cn,
                            int K, int D) {
  int k = blockIdx.x * blockDim.x + threadIdx.x;
  if (k < K) {
    float s = 0.f;
    const float* row = cb + (long)k * D;
    for (int d = 0; d < D; ++d) { float v = row[d]; s += v * v; }
    cn[k] = s;
  }
}

// ---------------------------------------------------------------------------
// 3-tap conv-as-GEMM with fused bias (+ optional ReLU, optional codebook
// gather on the A operand).  out[m][n] = act( sum_kk sum_i A(m+kk-1,i) *
// W[(kk*KC+i)][n] + bias[n] )
// Block: 256 threads (8 wave32), block tile 128M x 128N, wave tile 16M x 128N.
// ---------------------------------------------------------------------------
template <bool RELU, bool GATHER, int KC, int NN>
__global__ __launch_bounds__(256) void vqvae_conv3_wmma(
    const float* __restrict__ A,      // [M][KC] acts, or codebook [1024][KC]
    const int* __restrict__ gidx,     // GATHER: per-row codebook index
    const float* __restrict__ Wp,     // pair-packed [3*KC/2][NN*2]
    const float* __restrict__ bias,   // [NN]
    float* __restrict__ out) {        // [M][NN]
  // pair-packed tiles: element [kp][x] = {val(k=2kp, x), val(k=2kp+1, x)}
  __shared__ v2f A2[2][BK / 2][BM + 1];   // +1 v2f pad: de-alias half-waves
  __shared__ v2f B2[2][BK / 2][BN];

  const int tid  = threadIdx.x;
  const int lane = tid & 31;
  const int wave = tid >> 5;
  const int hi   = lane >> 4;   // upper half of wave
  const int l15  = lane & 15;
  const int m0   = blockIdx.x * BM;
  const int n0   = blockIdx.y * BN;
  const int wm   = wave * 16;

  // staging roles
  const int akp = tid & 7;      // A tile: k-pair 0..7
  const int ar0 = tid >> 3;     // A tile: row base 0..31 (4 rows, stride 32)
  const int bn  = tid & 127;    // B tile: n column
  const int bk0 = tid >> 7;     // B tile: k-pair base 0..1 (4 pairs, stride 2)

  constexpr int KSTEPS = KC / BK;
  constexpr int NSTEPS = 3 * KSTEPS;

  v8f acc[8] = {};
  v2f av[4], bv[4];

  auto loadA = [&](int s, int kb, v2f* av_) {
#pragma unroll
    for (int p = 0; p < 4; ++p) {
      int r  = ar0 + p * 32;
      int m  = m0 + r;
      int ts = (m & TMASK) + s;          // time index within its sequence
      v2f v  = {};
      if (ts >= 0 && ts < TSEQ) {
        int msrc = m + s;
        int arow = GATHER ? gidx[msrc] : msrc;
        const float* src = A + (long)(arow * KC + kb + akp * 2);
        v = *(const v2f*)src;
        if (p == 0) __builtin_prefetch(src + 2 * BK, 0, 0);  // 2 tiles ahead
      }
      av_[p] = v;
    }
  };
  auto loadB = [&](int kk, int kb, v2f* bv_) {
    // pair-row base of this (tap, k-block)
    int pr0 = (kk * KC + kb) >> 1;
    int col = n0 + bn;
    bool ok = (NN % BN == 0) || (col < NN);
#pragma unroll
    for (int p = 0; p < 4; ++p) {
      int kp = bk0 + p * 2;
      v2f v  = {};
      if (ok) v = *(const v2f*)(Wp + (long)(pr0 + kp) * (2 * NN) + 2 * col);
      bv_[p] = v;
    }
  };
  auto storeA = [&](int b, const v2f* av_) {
#pragma unroll
    for (int p = 0; p < 4; ++p) A2[b][akp][ar0 + p * 32] = av_[p];
  };
  auto storeB = [&](int b, const v2f* bv_) {
#pragma unroll
    for (int p = 0; p < 4; ++p) B2[b][bk0 + p * 2][bn] = bv_[p];
  };

  // prologue: step 0 = (tap 0, kb 0)
  loadA(-1, 0, av);
  loadB(0, 0, bv);
  storeA(0, av);
  storeB(0, bv);

  // incremental (tap, kb) of the *next* step: no division in the loop
  int kkN = 0, kbN = BK;
  if (kbN == KC) { kbN = 0; kkN = 1; }

  int buf = 0;
  for (int step = 0; step < NSTEPS; ++step) {
    __syncthreads();
    const bool more = (step + 1) < NSTEPS;
    if (more) { loadA(kkN - 1, kbN, av); loadB(kkN, kbN, bv); }

#pragma unroll
    for (int k4 = 0; k4 < 4; ++k4) {
      v2f af = A2[buf][k4 * 2 + hi][wm + l15];       // one ds_load_b64
      v2f bf[8];
#pragma unroll
      for (int nt = 0; nt < 8; ++nt)                 // batch all 8 frags
        bf[nt] = B2[buf][k4 * 2 + hi][nt * 16 + l15];
#pragma unroll
      for (int nt = 0; nt < 8; ++nt)                 // then the WMMA burst
        acc[nt] = wmma_f32x4(af, bf[nt], acc[nt]);
    }

    if (more) { storeA(buf ^ 1, av); storeB(buf ^ 1, bv); }
    buf ^= 1;
    kbN += BK;
    if (kbN == KC) { kbN = 0; ++kkN; }
  }

  // ---- epilogue: bias (+ReLU) and store ----
#pragma unroll
  for (int nt = 0; nt < 8; ++nt) {
    int col = n0 + nt * 16 + l15;
    if ((NN % BN == 0) || (col < NN)) {
      float bvv = bias[col];
#pragma unroll
      for (int r = 0; r < 8; ++r) {
        int row = m0 + wm + r + hi * 8;   // C layout: VGPR r -> M=r / M=r+8
        float v = acc[nt][r] + bvv;
        if (RELU) v = fmaxf(v, 0.f);
        out[(long)row * NN + col] = v;
      }
    }
  }
}

// ---------------------------------------------------------------------------
// VQ: argmin_n ( cnorm[n] - 2 * z[m]·cb[n] ) over all NCODE=1024 codes.
// Same GEMM core; each block owns 128 rows and loops the 8 N-chunks,
// reducing min+index across the 16 N-lanes with shfl_xor (first-min ties).
// ---------------------------------------------------------------------------
#define VQ_KC 512
#define VQ_N  1024

__global__ __launch_bounds__(256) void vqvae_vq_argmin_wmma(
    const float* __restrict__ Z,     // [M][VQ_KC]
    const float* __restrict__ CBp,   // pair-packed codebook^T [VQ_KC/2][2*VQ_N]
    const float* __restrict__ cnorm, // [VQ_N]
    int* __restrict__ idx_out) {     // [M]
  __shared__ v2f A2[2][BK / 2][BM + 1];
  __shared__ v2f B2[2][BK / 2][BN];

  const int tid  = threadIdx.x;
  const int lane = tid & 31;
  const int wave = tid >> 5;
  const int hi   = lane >> 4;
  const int l15  = lane & 15;
  const int m0   = blockIdx.x * BM;
  const int wm   = wave * 16;

  const int akp = tid & 7;
  const int ar0 = tid >> 3;
  const int bn  = tid & 127;
  const int bk0 = tid >> 7;

  constexpr int KSTEPS = VQ_KC / BK;

  float bestv[8];
  int   besti[8];
#pragma unroll
  for (int r = 0; r < 8; ++r) { bestv[r] = 3.4e38f; besti[r] = 0; }

  v2f av[4], bv[4];

  auto loadA = [&](int kb, v2f* av_) {
#pragma unroll
    for (int p = 0; p < 4; ++p) {
      int r = ar0 + p * 32;
      const float* src = Z + (long)((m0 + r) * VQ_KC + kb + akp * 2);
      av_[p] = *(const v2f*)src;
      if (p == 0) __builtin_prefetch(src + 2 * BK, 0, 0);
    }
  };
  auto loadB = [&](int nch, int kb, v2f* bv_) {
    int col = nch + bn;
#pragma unroll
    for (int p = 0; p < 4; ++p) {
      int kp = bk0 + p * 2;
      bv_[p] = *(const v2f*)(CBp + (long)((kb >> 1) + kp) * (2 * VQ_N) +
                             2 * col);
    }
  };
  auto storeA = [&](int b, const v2f* av_) {
#pragma unroll
    for (int p = 0; p < 4; ++p) A2[b][akp][ar0 + p * 32] = av_[p];
  };
  auto storeB = [&](int b, const v2f* bv_) {
#pragma unroll
    for (int p = 0; p < 4; ++p) B2[b][bk0 + p * 2][bn] = bv_[p];
  };

  for (int nch = 0; nch < VQ_N; nch += BN) {
    v8f acc[8] = {};

    loadA(0, av);
    loadB(nch, 0, bv);
    __syncthreads();           // protect buffers from previous chunk's readers
    storeA(0, av);
    storeB(0, bv);

    int buf = 0;
    for (int step = 0; step < KSTEPS; ++step) {
      __syncthreads();
      const bool more = (step + 1) < KSTEPS;
      if (more) { loadA((step + 1) * BK, av); loadB(nch, (step + 1) * BK, bv); }

#pragma unroll
      for (int k4 = 0; k4 < 4; ++k4) {
        v2f af = A2[buf][k4 * 2 + hi][wm + l15];
        v2f bf[8];
#pragma unroll
        for (int nt = 0; nt < 8; ++nt)
          bf[nt] = B2[buf][k4 * 2 + hi][nt * 16 + l15];
#pragma unroll
        for (int nt = 0; nt < 8; ++nt)
          acc[nt] = wmma_f32x4(af, bf[nt], acc[nt]);
      }

      if (more) { storeA(buf ^ 1, av); storeB(buf ^ 1, bv); }
      buf ^= 1;
    }

    // score + per-row min/argmin (16-lane half reduction, ties -> lowest idx)
#pragma unroll
    for (int r = 0; r < 8; ++r) {
#pragma unroll
      for (int nt = 0; nt < 8; ++nt) {
        int   n  = nch + nt * 16 + l15;
        float sv = cnorm[n] - 2.0f * acc[nt][r];
        int   bi = n;
#pragma unroll
        for (int msk = 1; msk < 16; msk <<= 1) {
          float ov = __shfl_xor(sv, msk, 32);
          int   oi = __shfl_xor(bi, msk, 32);
          if (ov < sv || (ov == sv && oi < bi)) { sv = ov; bi = oi; }
        }
        if (sv < bestv[r] || (sv == bestv[r] && bi < besti[r])) {
          bestv[r] = sv;
          besti[r] = bi;
        }
      }
    }
  }

  if (l15 == 0) {
#pragma unroll
    for (int r = 0; r < 8; ++r) {
      int row = m0 + wm + r + hi * 8;
      idx_out[row] = besti[r];
    }
  }
}

// ---------------------------------------------------------------------------
// Host-side orchestration
// ---------------------------------------------------------------------------
extern "C" void kernel_launch(void* const* d_in, const int* in_sizes, int n_in,
                              void* d_out, int out_size, void* d_ws,
                              size_t ws_size, hipStream_t stream) {
  (void)in_sizes; (void)n_in; (void)out_size; (void)ws_size;
  const int IN = 80, H = 512, D = 512, K = 1024;
  const int M = 16 * TSEQ;

  const float* mels     = (const float*)d_in[0];
  const float* enc_w1   = (const float*)d_in[1];
  const float* enc_b1   = (const float*)d_in[2];
  const float* enc_w2   = (const float*)d_in[3];
  const float* enc_b2   = (const float*)d_in[4];
  const float* codebook = (const float*)d_in[5];
  const float* dec_w1   = (const float*)d_in[6];
  const float* dec_b1   = (const float*)d_in[7];
  const float* dec_w2   = (const float*)d_in[8];
  const float* dec_b2   = (const float*)d_in[9];

  // workspace carve-up (floats)
  float* ws   = (float*)d_ws;
  float* wt1  = ws;                              // packed [3*IN/2][2*H]
  float* wt2  = wt1 + (size_t)3 * IN * H;        // packed [3*H/2][2*D]
  float* wt3  = wt2 + (size_t)3 * H * D;         // packed [3*D/2][2*H]
  float* wt4  = wt3 + (size_t)3 * D * H;         // packed [3*H/2][2*IN]
  float* cbp  = wt4 + (size_t)3 * H * IN;        // packed [D/2][2*K]
  float* cn   = cbp + (size_t)D * K;             // [K]
  float* act1 = cn + K;                          // [M][H] (reused for dec hid)
  float* act2 = act1 + (size_t)M * H;            // [M][D] (z)

  float* recon  = (float*)d_out;                    // [M][IN]
  int*   idxout = (int*)d_out + (size_t)M * IN;     // [M]

  // weight / codebook preprocessing
  int t;
  t = H * IN * 3;
  vqvae_wtrans<<<(t + 255) / 256, 256, 0, stream>>>(enc_w1, wt1, H, IN);
  t = D * H * 3;
  vqvae_wtrans<<<(t + 255) / 256, 256, 0, stream>>>(enc_w2, wt2, D, H);
  t = H * D * 3;
  vqvae_wtrans<<<(t + 255) / 256, 256, 0, stream>>>(dec_w1, wt3, H, D);
  t = IN * H * 3;
  vqvae_wtrans<<<(t + 255) / 256, 256, 0, stream>>>(dec_w2, wt4, IN, H);
  t = K * D;
  vqvae_cbT<<<(t + 255) / 256, 256, 0, stream>>>(codebook, cbp, K, D);
  vqvae_cnorm<<<(K + 255) / 256, 256, 0, stream>>>(codebook, cn, K, D);

  dim3 blk(256);
  // encoder
  vqvae_conv3_wmma<true, false, 80, 512>
      <<<dim3(M / BM, 512 / BN), blk, 0, stream>>>(mels, nullptr, wt1, enc_b1,
                                                   act1);
  vqvae_conv3_wmma<false, false, 512, 512>
      <<<dim3(M / BM, 512 / BN), blk, 0, stream>>>(act1, nullptr, wt2, enc_b2,
                                                   act2);
  // vector quantization (writes idx into d_out tail)
  vqvae_vq_argmin_wmma<<<dim3(M / BM), blk, 0, stream>>>(act2, cbp, cn, idxout);
  // decoder (conv1 fuses the codebook gather; act1 reused as hidden buffer)
  vqvae_conv3_wmma<true, true, 512, 512>
      <<<dim3(M / BM, 512 / BN), blk, 0, stream>>>(codebook, idxout, wt3,
                                                   dec_b1, act1);
  vqvae_conv3_wmma<false, false, 512, 80>
      <<<dim3(M / BM, 1), blk, 0, stream>>>(act1, nullptr, wt4, dec_b2, recon);
}